// GraphConv_and_gather_15676630631151
// MI455X (gfx1250) — compile-verified
//
#include <hip/hip_runtime.h>
#include <hip/hip_bf16.h>
#include <math.h>

// ---------------------------------------------------------------------------
// GraphConv + GraphGather fused pipeline for MI455X (gfx1250, wave32, WMMA)
//
// - GEMMs on V_WMMA_F32_16X16X4_F32 (fp32 exact; 31.5 GFLOP is free, the
//   workload is ~350MB HBM traffic => ~15us roofline at 23.3 TB/s).
// - Weight slabs staged LDS-side by the Tensor Data Mover (tensor_load_to_lds,
//   TENSORcnt) so the vector pipes stay on the neighbor-gather stream.
// - Bias + layernorm + segment_sum + segment_max fused into the GEMM epilogue
//   so activated/gathered never round-trip HBM.
// ---------------------------------------------------------------------------

#define MAX_DEG   10
#define F         128
#define BATCHM    128
#define DEG_COUNT 30000
#define N_ATOMS   ((MAX_DEG + 1) * DEG_COUNT)   // 330000
#define TILES_PER_DEG (DEG_COUNT / 16)          // 1875
#define WAVES_PER_BLK 4
#define EPSF      1e-5f
#define WROW      129                           // LDS row stride (TDM pad: +1 dword)

typedef float    v8f __attribute__((ext_vector_type(8)));
typedef float    v2f __attribute__((ext_vector_type(2)));
typedef unsigned v4u __attribute__((ext_vector_type(4)));
typedef unsigned v8u __attribute__((ext_vector_type(8)));

struct KParams {
    const float* atoms;      // (330000,128)
    const int*   mem;        // (330000,)
    const int*   adj[10];    // adj[d-1]: (30000, d)
    const float* W;          // (32,128,128)
    const float* b;          // (32,128)
    const float* gamma;      // (128,)
    const float* beta;       // (128,)
    float*       xn;         // out: (330000,128)
    float*       asum;       // ws:  (128,128) segment sums
    unsigned*    gmax;       // ws:  (128,128) encoded segment max
};

// Order-preserving float->uint encoding so segment_max can use atomicMax(u32).
__device__ __forceinline__ unsigned encf(float f) {
    unsigned u = __float_as_uint(f);
    return (u & 0x80000000u) ? ~u : (u | 0x80000000u);
}
__device__ __forceinline__ float decf(unsigned e) {
    unsigned u = (e & 0x80000000u) ? (e ^ 0x80000000u) : ~e;
    return __uint_as_float(u);
}

// Low 32 bits of a generic LDS pointer = LDS byte offset (shared aperture).
// Laundered through asm so the compiler must assume the TDM writes the array.
__device__ __forceinline__ unsigned lds_byte_offset(const void* p) {
    unsigned long long a = (unsigned long long)(uintptr_t)p;
    asm volatile("" : "+s"(a));
    return (unsigned)a;
}

// Issue one TDM 2D tile load: 32 rows x 128 f32 from row-major W into LDS,
// with 1-dword padding after every 128 dwords (row stride -> 129 dwords so
// consecutive K-rows land in different LDS banks).
// D# packing per CDNA5 ISA ch.8 (group0: flags/lds/global/type, group1: dims).
__device__ __forceinline__ void tdm_load_w_slab(const float* gsrc, unsigned lds_off) {
    unsigned long long ga = (unsigned long long)(uintptr_t)gsrc;
    v4u g0;
    g0.x = 1u;                                   // count=1 (user descriptor)
    g0.y = lds_off;                              // lds_addr (bytes)
    g0.z = (unsigned)ga;                         // global_addr[31:0]
    g0.w = (unsigned)(ga >> 32) | (2u << 30);    // global_addr[56:32] | type=2
    v8u g1;
    g1[0] = (2u << 16)          // data_size = 4B
          | (1u << 20)          // pad_enable
          | (6u << 22)          // pad_interval code 6 = every 128 dwords
          | (0u << 25);         // pad_amount  code 0 = 1 dword
    g1[1] = 128u << 16;         // tensor_dim0 = 128      (bits 63:48)
    g1[2] = 32u  << 16;         // tensor_dim1 = 32       (bits 95:80)
    g1[3] = 128u << 16;         // tile_dim0   = 128      (bits 127:112)
    g1[4] = 32u;                // tile_dim1   = 32       (bits 143:128)
    g1[5] = 128u;               // tensor_dim0_stride=128 (bits 191:160)
    g1[6] = 0u;
    g1[7] = 0u;
    v4u gz = (v4u)(0u);
    asm volatile("tensor_load_to_lds %0, %1, %2, %3"
                 :: "s"(g0), "s"(g1), "s"(gz), "s"(gz)
                 : "memory");
}

__global__ void graphconv_init(float* asum, unsigned* gmax) {
    int i = blockIdx.x * blockDim.x + threadIdx.x;
    if (i < BATCHM * F) {
        asum[i] = 0.0f;
        gmax[i] = 0u;   // below any encoded float value that can occur
    }
}

__global__ __launch_bounds__(128)
void graphconv_main(KParams p) {
    // Row-major W slabs (TDM destination), 32 K-rows x 129 dwords each.
    __shared__ float wstage[3][32 * WROW];   // ~48.4 KB

    const int tid  = threadIdx.x;
    const int wave = tid >> 5;
    const int lane = tid & 31;
    const int h    = lane >> 4;     // half-wave: K phase (A/B) / row group (C)
    const int c    = lane & 15;     // column phase within a 16-wide N tile
    const int d    = blockIdx.y;    // degree bucket 0..10

    const int tile     = blockIdx.x * WAVES_PER_BLK + wave;
    const bool active  = (tile < TILES_PER_DEG);
    const int rowbase  = tile * 16;                 // row in bucket
    const int growbase = d * DEG_COUNT + rowbase;   // global atom row

    // Weight indices matching the reference consumption order.
    const int i0 = (d >= 1) ? 2 * (d - 1)  : 20;    // neigh-rel (d0: unused)
    const int i1 = (d >= 1) ? 2 * d - 1    : 20;    // self-rel
    const int ig = (d >= 1) ? 20 + d       : 31;    // gather

    const int m = c;                                // strip row for this lane pair

    const unsigned lds0 = lds_byte_offset(&wstage[0][0]);
    const unsigned lds1 = lds_byte_offset(&wstage[1][0]);
    const unsigned lds2 = lds_byte_offset(&wstage[2][0]);

    v8f c_rel[8] = {};   // rel accumulators, 8 N-tiles
    v8f c_gth[8] = {};   // gather accumulators

    for (int slab = 0; slab < 4; ++slab) {          // K = 128 in 4 slabs of 32
        __syncthreads();                            // prior slab fully consumed
        if (wave == 0) {
            // TDM stages the three 16KB W slabs; vector pipes stay free.
            tdm_load_w_slab(p.W + ((size_t)i0 * F + (size_t)slab * 32) * F, lds0);
            tdm_load_w_slab(p.W + ((size_t)i1 * F + (size_t)slab * 32) * F, lds1);
            tdm_load_w_slab(p.W + ((size_t)ig * F + (size_t)slab * 32) * F, lds2);
            __builtin_amdgcn_s_wait_tensorcnt(0);
        } else if (wave == 1 && slab < 3) {
            // Warm GL2 for next slab's TDM reads (global_prefetch_b8).
            const int nk = 4 * (slab + 1) * 8;
            __builtin_prefetch(p.W + ((size_t)i0 * F + nk) * F + (size_t)lane * F, 0, 0);
            __builtin_prefetch(p.W + ((size_t)i1 * F + nk) * F + (size_t)lane * F, 0, 0);
            __builtin_prefetch(p.W + ((size_t)ig * F + nk) * F + (size_t)lane * F, 0, 0);
        }
        __syncthreads();                            // slab visible to all waves

        if (active) {
            // --- A fragments for this slab (registers) ---
            v2f a_self[8], a_ngh[8];
            const float* selfrow = p.atoms + (size_t)(growbase + m) * F;
            #pragma unroll
            for (int ks = 0; ks < 8; ++ks)
                a_self[ks] = *(const v2f*)(selfrow + 4 * (slab * 8 + ks) + 2 * h);

            if (d >= 1) {
                #pragma unroll
                for (int ks = 0; ks < 8; ++ks) { a_ngh[ks].x = 0.f; a_ngh[ks].y = 0.f; }
                const int* adjrow = p.adj[d - 1] + (size_t)(rowbase + m) * d;
                for (int j = 0; j < d; ++j) {                 // neighbor gather+sum
                    const float* nr = p.atoms + (size_t)adjrow[j] * F;
                    #pragma unroll
                    for (int ks = 0; ks < 8; ++ks) {
                        v2f v = *(const v2f*)(nr + 4 * (slab * 8 + ks) + 2 * h);
                        a_ngh[ks].x += v.x;
                        a_ngh[ks].y += v.y;
                    }
                }
            }

            // --- WMMA: C += A * B over 8 K-steps x 8 N-tiles ---
            // B fragment: lane (h,c) needs rows 4k+2h, 4k+2h+1 at col n*16+c.
            #pragma unroll
            for (int n = 0; n < 8; ++n) {
                if (d >= 1) {
                    #pragma unroll
                    for (int ks = 0; ks < 8; ++ks) {
                        v2f bf;
                        bf.x = wstage[0][(4 * ks + 2 * h)     * WROW + n * 16 + c];
                        bf.y = wstage[0][(4 * ks + 2 * h + 1) * WROW + n * 16 + c];
                        c_rel[n] = __builtin_amdgcn_wmma_f32_16x16x4_f32(
                            false, a_ngh[ks], false, bf, (short)0, c_rel[n], false, false);
                    }
                }
                #pragma unroll
                for (int ks = 0; ks < 8; ++ks) {
                    v2f bf;
                    bf.x = wstage[1][(4 * ks + 2 * h)     * WROW + n * 16 + c];
                    bf.y = wstage[1][(4 * ks + 2 * h + 1) * WROW + n * 16 + c];
                    c_rel[n] = __builtin_amdgcn_wmma_f32_16x16x4_f32(
                        false, a_self[ks], false, bf, (short)0, c_rel[n], false, false);
                }
                #pragma unroll
                for (int ks = 0; ks < 8; ++ks) {
                    v2f bf;
                    bf.x = wstage[2][(4 * ks + 2 * h)     * WROW + n * 16 + c];
                    bf.y = wstage[2][(4 * ks + 2 * h + 1) * WROW + n * 16 + c];
                    c_gth[n] = __builtin_amdgcn_wmma_f32_16x16x4_f32(
                        false, a_self[ks], false, bf, (short)0, c_gth[n], false, false);
                }
            }
        }
    }

    if (!active) return;

    // --- Fused epilogue: bias + layernorm + segment_sum + segment_max ---
    float brel[8], bgth[8], gam[8], bet[8];
    #pragma unroll
    for (int n = 0; n < 8; ++n) {
        const int col = n * 16 + c;
        float bb = p.b[i1 * F + col];
        if (d >= 1) bb += p.b[i0 * F + col];
        brel[n] = bb;
        bgth[n] = p.b[ig * F + col];
        gam[n]  = p.gamma[col];
        bet[n]  = p.beta[col];
    }
    #pragma unroll
    for (int n = 0; n < 8; ++n)
        #pragma unroll
        for (int j = 0; j < 8; ++j) {
            c_rel[n][j] += brel[n];
            c_gth[n][j] += bgth[n];
        }

    // Row r = j + 8*h lives in element j of each C fragment, spread over the
    // 16 lanes of this half-wave (col = n*16 + lane%16). Reduce within the
    // 16-lane group via xor shuffles (masks 1,2,4,8 stay inside the group).
    float mean[8], rstd[8];
    #pragma unroll
    for (int j = 0; j < 8; ++j) {
        float s = 0.f;
        #pragma unroll
        for (int n = 0; n < 8; ++n) s += c_rel[n][j];
        s += __shfl_xor(s, 1, 32);
        s += __shfl_xor(s, 2, 32);
        s += __shfl_xor(s, 4, 32);
        s += __shfl_xor(s, 8, 32);
        mean[j] = s * (1.0f / 128.0f);
    }
    #pragma unroll
    for (int j = 0; j < 8; ++j) {
        float s = 0.f;
        #pragma unroll
        for (int n = 0; n < 8; ++n) {
            const float dv = c_rel[n][j] - mean[j];
            s += dv * dv;
        }
        s += __shfl_xor(s, 1, 32);
        s += __shfl_xor(s, 2, 32);
        s += __shfl_xor(s, 4, 32);
        s += __shfl_xor(s, 8, 32);
        const float var = s * (1.0f / 128.0f);
        rstd[j] = 1.0f / (sqrtf(var + EPSF) + EPSF);
    }

    #pragma unroll
    for (int j = 0; j < 8; ++j) {
        const int grow = growbase + j + 8 * h;
        const int mol  = p.mem[grow];
        float*    xrow = p.xn   + (size_t)grow * F;
        float*    arow = p.asum + (size_t)mol * F;
        unsigned* grw  = p.gmax + (size_t)mol * F;
        #pragma unroll
        for (int n = 0; n < 8; ++n) {
            const int col = n * 16 + c;
            const float xv = gam[n] * ((c_rel[n][j] - mean[j]) * rstd[j]) + bet[n];
            xrow[col] = xv;
            atomicMax(&grw[col], encf(xv));
            atomicAdd(&arow[col], c_gth[n][j]);
        }
    }
}

__global__ __launch_bounds__(128)
void graphconv_finalize(const float* asum, const unsigned* gmax,
                        const float* gamma, const float* beta,
                        float* yn, float* g) {
    __shared__ float red[128];
    const int row = blockIdx.x;
    const int t   = threadIdx.x;
    const float v = asum[row * F + t];

    red[t] = v;
    __syncthreads();
    #pragma unroll
    for (int s = 64; s > 0; s >>= 1) {
        if (t < s) red[t] += red[t + s];
        __syncthreads();
    }
    const float m = red[0] * (1.0f / 128.0f);
    __syncthreads();

    const float dv = v - m;
    red[t] = dv * dv;
    __syncthreads();
    #pragma unroll
    for (int s = 64; s > 0; s >>= 1) {
        if (t < s) red[t] += red[t + s];
        __syncthreads();
    }
    const float var  = red[0] * (1.0f / 128.0f);
    const float rstd = 1.0f / (sqrtf(var + EPSF) + EPSF);

    yn[row * F + t] = gamma[t] * ((v - m) * rstd) + beta[t];
    g [row * F + t] = decf(gmax[row * F + t]);
}

extern "C" void kernel_launch(void* const* d_in, const int* in_sizes, int n_in,
                              void* d_out, int out_size, void* d_ws, size_t ws_size,
                              hipStream_t stream) {
    (void)in_sizes; (void)n_in; (void)out_size; (void)ws_size;

    KParams p;
    p.atoms = (const float*)d_in[0];
    // d_in[1] = deg_slice (static, unused)
    p.mem   = (const int*)d_in[2];
    for (int d = 1; d <= 10; ++d) p.adj[d - 1] = (const int*)d_in[2 + d];
    p.W     = (const float*)d_in[13];
    p.b     = (const float*)d_in[14];
    p.gamma = (const float*)d_in[15];
    p.beta  = (const float*)d_in[16];

    float* out = (float*)d_out;
    p.xn = out;                                              // (330000,128)
    float* yn = out + (size_t)N_ATOMS * F;                   // (128,128)
    float* g  = yn + (size_t)BATCHM * F;                     // (128,128)

    p.asum = (float*)d_ws;                                   // 64 KB
    p.gmax = (unsigned*)((char*)d_ws + (size_t)BATCHM * F * sizeof(float));

    graphconv_init<<<dim3((BATCHM * F + 255) / 256), 256, 0, stream>>>(p.asum, p.gmax);

    const int gx = (TILES_PER_DEG + WAVES_PER_BLK - 1) / WAVES_PER_BLK;  // 469
    graphconv_main<<<dim3(gx, MAX_DEG + 1), 128, 0, stream>>>(p);

    graphconv_finalize<<<dim3(BATCHM), 128, 0, stream>>>(p.asum, p.gmax,
                                                         p.gamma, p.beta, yn, g);
}